// Encoder_41575283425665
// MI455X (gfx1250) — compile-verified
//
#include <hip/hip_runtime.h>

// ---------------------------------------------------------------------------
// 2-layer GraphSAGE (mean aggr) for MI455X / gfx1250.
//   N=50000 nodes, E=800000 edges, D_IN=96, D_HID=D_OUT=128.
//
// Design: L2-resident CSR aggregation (no float atomics; the naive
// scatter-add would need ~179M L2 atomic RMWs and be atomic-throughput
// bound) + fused SAGE layers through V_WMMA_F32_16X16X4_F32 (exact fp32,
// GEMM flops are negligible vs. the gather phase).
// ---------------------------------------------------------------------------

typedef __attribute__((ext_vector_type(2))) float v2f;
typedef __attribute__((ext_vector_type(8))) float v8f;

#define N_NODES 50000
#define N_EDGES 800000
#define D_IN    96
#define D_HID   128
#define D_OUT   128

// ---------------- zero ints ----------------
__global__ void zero_int_kernel(int* __restrict__ p, int n) {
    int t = blockIdx.x * blockDim.x + threadIdx.x;
    if (t < n) p[t] = 0;
}

// ---------------- in-degree counts ----------------
__global__ void count_kernel(const int* __restrict__ dst,
                             int* __restrict__ cnt, int n_edges) {
    int t = blockIdx.x * blockDim.x + threadIdx.x;
    if (t < n_edges) atomicAdd(&cnt[dst[t]], 1);
}

// ---------------- single-block exclusive scan (N=50000) ----------------
__global__ __launch_bounds__(1024) void scan_kernel(const int* __restrict__ cnt,
                                                    int* __restrict__ rs, int n) {
    __shared__ int buf[1024];
    __shared__ int s_carry;
    if (threadIdx.x == 0) s_carry = 0;
    __syncthreads();
    for (int base = 0; base < n; base += 1024) {
        const int i = base + threadIdx.x;
        const int v = (i < n) ? cnt[i] : 0;
        int acc = v;
        buf[threadIdx.x] = acc;
        __syncthreads();
        for (int off = 1; off < 1024; off <<= 1) {
            const int t = (threadIdx.x >= (unsigned)off) ? buf[threadIdx.x - off] : 0;
            __syncthreads();
            acc += t;
            buf[threadIdx.x] = acc;
            __syncthreads();
        }
        const int carry = s_carry;
        if (i < n) rs[i] = carry + acc - v;      // exclusive
        __syncthreads();
        if (threadIdx.x == 1023) s_carry = carry + acc;
        __syncthreads();
    }
    if (threadIdx.x == 0) rs[n] = s_carry;       // == N_EDGES
}

// ---------------- CSR fill: group src ids by dst ----------------
__global__ void fill_kernel(const int* __restrict__ src,
                            const int* __restrict__ dst,
                            const int* __restrict__ rs,
                            int* __restrict__ cursor,
                            int* __restrict__ eidx, int n_edges) {
    int t = blockIdx.x * blockDim.x + threadIdx.x;
    if (t >= n_edges) return;
    const int d = dst[t];
    const int p = atomicAdd(&cursor[d], 1);
    eidx[rs[d] + p] = src[t];
}

// ---------------- per-node insertion sort (deterministic sum order) -------
__global__ void sort_segments_kernel(const int* __restrict__ rs,
                                     int* __restrict__ eidx, int n) {
    int i = blockIdx.x * blockDim.x + threadIdx.x;
    if (i >= n) return;
    const int beg = rs[i], end = rs[i + 1];
    for (int a = beg + 1; a < end; ++a) {
        const int v = eidx[a];
        int b = a - 1;
        while (b >= beg && eidx[b] > v) { eidx[b + 1] = eidx[b]; --b; }
        eidx[b + 1] = v;
    }
}

// ---------------- gather mean-aggregate: out[i] = mean_{j->i} feat[j] -----
// One block per node, D threads (feature dims); each edge row read is a
// fully-coalesced D*4B burst (L2-resident: feat arrays are 19-26 MB << 192MB).
template <int D>
__global__ void aggregate_mean_kernel(const float* __restrict__ feat,
                                      const int* __restrict__ rs,
                                      const int* __restrict__ eidx,
                                      float* __restrict__ out) {
    const int node = blockIdx.x;
    const int f = threadIdx.x;
    const int beg = rs[node], end = rs[node + 1];
    float sum = 0.0f;
    for (int j = beg; j < end; ++j) {
        const int s = eidx[j];                   // scalar-uniform per block
        sum += feat[(size_t)s * D + f];
    }
    const float d = (float)(end - beg);
    out[(size_t)node * D + f] = sum / fmaxf(d, 1.0f);   // exact IEEE, ref-match
}

// ---------------- fused SAGE layer via f32 WMMA ----------------
// out[M x 128] = act( agg @ Wl + bias + xin @ Wr ),  agg already mean'd.
// Block = 256 threads = 8 waves; block b -> 16-row tile, wave w -> 16-col
// tile. V_WMMA_F32_16X16X4_F32 operand layouts per CDNA5 ISA 7.12.2.
template <int K, bool RELU>
__global__ __launch_bounds__(256) void sage_wmma_kernel(
        const float* __restrict__ agg,   // [M x K]  (mean aggregate)
        const float* __restrict__ xin,   // [M x K]  (self features)
        const float* __restrict__ Wl,    // [K x 128] row-major
        const float* __restrict__ Wr,    // [K x 128] row-major
        const float* __restrict__ bias,  // [128]
        float* __restrict__ out) {       // [M x 128]
    const int lane = threadIdx.x & 31;
    const int wave = threadIdx.x >> 5;       // n-tile 0..7
    const int row0 = blockIdx.x * 16;        // m-tile base
    const int n0   = wave * 16;

    const int half = lane >> 4;              // 0 | 1
    const int l15  = lane & 15;
    const int arow = row0 + l15;             // A row for this lane
    const int kA   = 2 * half;               // A k-pair offset (0 | 2)
    const int kB0  = 2 * half;               // B vgpr0 k offset (0 | 2)
    const int kB1  = kB0 + 1;                // B vgpr1 k offset (1 | 3)
    const int bcol = n0 + l15;               // B/C/D column

    v8f acc = {};

    const float* aptr = agg + (size_t)arow * K;
    #pragma unroll
    for (int k0 = 0; k0 < K; k0 += 4) {
        v2f a, b;
        a.x = aptr[k0 + kA + 0];
        a.y = aptr[k0 + kA + 1];
        b.x = Wl[(k0 + kB0) * 128 + bcol];
        b.y = Wl[(k0 + kB1) * 128 + bcol];
        acc = __builtin_amdgcn_wmma_f32_16x16x4_f32(
                false, a, false, b, (short)0, acc, false, false);
    }

    const float* xptr = xin + (size_t)arow * K;
    #pragma unroll
    for (int k0 = 0; k0 < K; k0 += 4) {
        v2f a, b;
        a.x = xptr[k0 + kA + 0];
        a.y = xptr[k0 + kA + 1];
        b.x = Wr[(k0 + kB0) * 128 + bcol];
        b.y = Wr[(k0 + kB1) * 128 + bcol];
        acc = __builtin_amdgcn_wmma_f32_16x16x4_f32(
                false, a, false, b, (short)0, acc, false, false);
    }

    const float bb = bias[bcol];
    #pragma unroll
    for (int v = 0; v < 8; ++v) {
        const int r = row0 + v + 8 * half;
        float val = acc[v] + bb;
        if (RELU) val = fmaxf(val, 0.0f);
        out[(size_t)r * 128 + bcol] = val;
    }
}

// ---------------------------------------------------------------------------
extern "C" void kernel_launch(void* const* d_in, const int* in_sizes, int n_in,
                              void* d_out, int out_size, void* d_ws, size_t ws_size,
                              hipStream_t stream) {
    const float* x   = (const float*)d_in[0];   // [N, 96]
    const int*   ei  = (const int*)d_in[1];     // [2, E]
    const float* W1l = (const float*)d_in[2];   // [96, 128]
    const float* b1  = (const float*)d_in[3];   // [128]
    const float* W1r = (const float*)d_in[4];   // [96, 128]
    const float* W2l = (const float*)d_in[5];   // [128, 128]
    const float* b2  = (const float*)d_in[6];   // [128]
    const float* W2r = (const float*)d_in[7];   // [128, 128]
    float* out = (float*)d_out;                 // [N, 128]

    const int* src = ei;
    const int* dst = ei + N_EDGES;

    // Workspace layout: int region, then 16B-aligned float region.
    int* cnt       = (int*)d_ws;                 // N      (counts)
    int* cursor    = cnt + N_NODES;              // N      (fill cursors)
    int* row_start = cursor + N_NODES;           // N+1
    int* eidx      = row_start + (N_NODES + 1);  // E      (src ids grouped by dst)

    size_t int_words = 2ull * N_NODES + (N_NODES + 1) + N_EDGES;
    int_words = (int_words + 3) & ~(size_t)3;    // 16B align float region
    float* agg1 = (float*)d_ws + int_words;              // N*96
    float* h    = agg1 + (size_t)N_NODES * D_IN;         // N*128
    float* agg2 = h + (size_t)N_NODES * D_HID;           // N*128

    const int BS = 256;

    // 1) zero cnt + cursor (contiguous)
    zero_int_kernel<<<(2 * N_NODES + BS - 1) / BS, BS, 0, stream>>>(cnt, 2 * N_NODES);

    // 2) CSR build: count -> scan -> fill -> per-node sort (determinism)
    count_kernel<<<(N_EDGES + BS - 1) / BS, BS, 0, stream>>>(dst, cnt, N_EDGES);
    scan_kernel<<<1, 1024, 0, stream>>>(cnt, row_start, N_NODES);
    fill_kernel<<<(N_EDGES + BS - 1) / BS, BS, 0, stream>>>(
        src, dst, row_start, cursor, eidx, N_EDGES);
    sort_segments_kernel<<<(N_NODES + BS - 1) / BS, BS, 0, stream>>>(
        row_start, eidx, N_NODES);

    // 3) layer 1: agg1 = mean_{j->i} x[j];  h = relu(agg1@W1l + b1 + x@W1r)
    aggregate_mean_kernel<D_IN><<<N_NODES, D_IN, 0, stream>>>(x, row_start, eidx, agg1);
    sage_wmma_kernel<D_IN, true><<<N_NODES / 16, BS, 0, stream>>>(
        agg1, x, W1l, W1r, b1, h);

    // 4) layer 2: agg2 = mean_{j->i} h[j];  out = agg2@W2l + b2 + h@W2r
    aggregate_mean_kernel<D_HID><<<N_NODES, D_HID, 0, stream>>>(h, row_start, eidx, agg2);
    sage_wmma_kernel<D_HID, false><<<N_NODES / 16, BS, 0, stream>>>(
        agg2, h, W2l, W2r, b2, out);
}